// LossAdjacency_31885837205881
// MI455X (gfx1250) — compile-verified
//
#include <hip/hip_runtime.h>
#include <math.h>

// Problem constants (match reference)
#define N_VERT 12288
#define C_CLS  32
#define TILE   16
#define CHUNKS 16                               // j-dimension chunks (grid.y)
#define CHUNK_W (N_VERT / CHUNKS)               // 768 columns per chunk
#define TILES_PER_CHUNK (CHUNK_W / TILE)        // 48 tiles per chunk
#define WAVES_PER_BLOCK 8
#define TILES_PER_WAVE (TILES_PER_CHUNK / WAVES_PER_BLOCK)  // 6

#define LOG2E  1.4426950408889634f
#define LN2    0.6931471805599453f

typedef float v2f __attribute__((ext_vector_type(2)));
typedef float v8f __attribute__((ext_vector_type(8)));

// ---------------------------------------------------------------------------
// Kernel 1: p[i] = log_softmax(pred[i,:])[gt[i]]   (one wave32 per row, C==32)
// ---------------------------------------------------------------------------
__global__ void __launch_bounds__(256)
logp_gather_kernel(const float* __restrict__ pred,
                   const int*   __restrict__ gt,
                   float*       __restrict__ p_out) {
    const int lane = threadIdx.x & 31;
    const int wave = threadIdx.x >> 5;
    const int row  = blockIdx.x * WAVES_PER_BLOCK + wave;
    if (row >= N_VERT) return;

    float x = pred[row * C_CLS + lane];          // lane == class index (C==32)

    float m = x;                                  // wave max
    #pragma unroll
    for (int off = 16; off >= 1; off >>= 1)
        m = fmaxf(m, __shfl_xor(m, off, 32));

    float s = __builtin_amdgcn_exp2f((x - m) * LOG2E);  // wave sum of exp
    #pragma unroll
    for (int off = 16; off >= 1; off >>= 1)
        s += __shfl_xor(s, off, 32);

    const int g = gt[row];
    const float xg = __shfl(x, g, 32);            // broadcast pred[row, g]
    if (lane == 0)
        p_out[row] = xg - (m + LN2 * __builtin_amdgcn_logf(s));
}

// ---------------------------------------------------------------------------
// Kernel 2: pairwise tile loop. d2 via V_WMMA_F32_16X16X4_F32:
//   A[i,:] = (-2vx, -2vy, -2vz, sq_i), B[:,j] = (vx, vy, vz, 1), C = sq_j
//   => D = sq_i + sq_j - 2 v_i.v_j
// adj streamed non-temporally (single-use 604 MB stream).
// ---------------------------------------------------------------------------
__global__ void __launch_bounds__(256)
pair_loss_kernel(const float* __restrict__ verts,
                 const float* __restrict__ adj,
                 const float* __restrict__ p,
                 float*       __restrict__ partAcc,
                 float*       __restrict__ partAdj) {
    const int i0    = blockIdx.x * TILE;
    const int chunk = blockIdx.y;
    const int jbase = chunk * CHUNK_W;
    const int wave  = threadIdx.x >> 5;
    const int lane  = threadIdx.x & 31;
    const int half  = lane >> 4;                  // 0: lanes 0-15, 1: lanes 16-31
    const int li    = lane & 15;

    // --- A matrix (fixed for this block), ISA f32 16x4 layout:
    // lanes 0-15: M=lane, K=0..1 ; lanes 16-31: M=lane-16, K=2..3
    const int arow = i0 + li;
    const float avx = verts[arow * 3 + 0];
    const float avy = verts[arow * 3 + 1];
    const float avz = verts[arow * 3 + 2];
    const float sqi = avx * avx + avy * avy + avz * avz;
    v2f A;
    A.x = half ? (-2.0f * avz) : (-2.0f * avx);   // K=2 : K=0
    A.y = half ? sqi           : (-2.0f * avy);   // K=3 : K=1

    // p_i for the C/D rows this lane touches (row m = r + 8*half)
    float pim[8];
    #pragma unroll
    for (int r = 0; r < 8; ++r) pim[r] = p[i0 + r + 8 * half];

    float acc[8], ads[8];
    #pragma unroll
    for (int r = 0; r < 8; ++r) { acc[r] = 0.0f; ads[r] = 0.0f; }

    for (int t = 0; t < TILES_PER_WAVE; ++t) {
        const int j0  = jbase + (t * WAVES_PER_BLOCK + wave) * TILE;
        const int col = j0 + li;

        // --- B matrix, ISA f32 4x16 layout:
        // VGPR0: K=0 (lanes 0-15) / K=2 (lanes 16-31); VGPR1: K=1 / K=3
        const float bvx = verts[col * 3 + 0];
        const float bvy = verts[col * 3 + 1];
        const float bvz = verts[col * 3 + 2];
        const float sqj = bvx * bvx + bvy * bvy + bvz * bvz;
        v2f B;
        B.x = half ? bvz  : bvx;
        B.y = half ? 1.0f : bvy;

        v8f Cm;
        #pragma unroll
        for (int r = 0; r < 8; ++r) Cm[r] = sqj;  // C[m][n] = sq_j (col of lane)

        // D = A x B + C  (one 16x16 tile of squared distances)
        v8f D = __builtin_amdgcn_wmma_f32_16x16x4_f32(
            false, A, false, B, (short)0, Cm, false, false);

        const float pj = p[col];

        #pragma unroll
        for (int r = 0; r < 8; ++r) {
            // C/D layout: element r -> row i0 + r + 8*half, col j0 + li
            const float av = __builtin_nontemporal_load(
                adj + (size_t)(i0 + r + 8 * half) * N_VERT + col);
            const float d2  = D[r];
            // hardware sqrt / exp2: single v_sqrt_f32 + v_exp_f32 per element
            const float euc = __builtin_amdgcn_sqrtf(fmaxf(d2, 0.0f));
            const float w   = __builtin_amdgcn_exp2f(-10.0f * LOG2E * euc); // exp(-euc/0.1)
            const float cb  = w * fabsf(pim[r] - pj);
            acc[r] = fmaf(cb, av, acc[r]);
            ads[r] += av;
        }
    }

    // Reduce across the 16 lanes of each half-group (xor 8..1 stays in-group)
    #pragma unroll
    for (int r = 0; r < 8; ++r) {
        #pragma unroll
        for (int off = 8; off >= 1; off >>= 1) {
            acc[r] += __shfl_xor(acc[r], off, 32);
            ads[r] += __shfl_xor(ads[r], off, 32);
        }
    }

    // Cross-wave reduction in LDS, then one deterministic write per row
    __shared__ float sAcc[WAVES_PER_BLOCK][TILE];
    __shared__ float sAdj[WAVES_PER_BLOCK][TILE];
    if (li == 0) {
        #pragma unroll
        for (int r = 0; r < 8; ++r) {
            sAcc[wave][r + 8 * half] = acc[r];
            sAdj[wave][r + 8 * half] = ads[r];
        }
    }
    __syncthreads();
    if (threadIdx.x < TILE) {
        float a = 0.0f, b = 0.0f;
        #pragma unroll
        for (int w = 0; w < WAVES_PER_BLOCK; ++w) {
            a += sAcc[w][threadIdx.x];
            b += sAdj[w][threadIdx.x];
        }
        partAcc[(size_t)chunk * N_VERT + i0 + threadIdx.x] = a;
        partAdj[(size_t)chunk * N_VERT + i0 + threadIdx.x] = b;
    }
}

// ---------------------------------------------------------------------------
// Kernel 3: fold chunk partials per row, apply adj_num guard, emit scalar.
// Single block => fixed summation order => deterministic.
// ---------------------------------------------------------------------------
__global__ void __launch_bounds__(256)
final_reduce_kernel(const float* __restrict__ partAcc,
                    const float* __restrict__ partAdj,
                    float*       __restrict__ out) {
    float local = 0.0f;
    for (int i = threadIdx.x; i < N_VERT; i += 256) {
        float a = 0.0f, b = 0.0f;
        #pragma unroll
        for (int c = 0; c < CHUNKS; ++c) {
            a += partAcc[(size_t)c * N_VERT + i];
            b += partAdj[(size_t)c * N_VERT + i];
        }
        b = (b == 0.0f) ? 1.0f : b;
        local += a / b;
    }
    #pragma unroll
    for (int off = 16; off >= 1; off >>= 1)
        local += __shfl_xor(local, off, 32);

    __shared__ float sw[WAVES_PER_BLOCK];
    const int lane = threadIdx.x & 31, wave = threadIdx.x >> 5;
    if (lane == 0) sw[wave] = local;
    __syncthreads();
    if (threadIdx.x == 0) {
        float t = 0.0f;
        #pragma unroll
        for (int w = 0; w < WAVES_PER_BLOCK; ++w) t += sw[w];
        out[0] = t / (float)N_VERT;
    }
}

// ---------------------------------------------------------------------------
extern "C" void kernel_launch(void* const* d_in, const int* in_sizes, int n_in,
                              void* d_out, int out_size, void* d_ws, size_t ws_size,
                              hipStream_t stream) {
    const float* pred  = (const float*)d_in[0];   // (N, 32) f32
    const int*   gt    = (const int*)d_in[1];     // (N, 1) int
    const float* verts = (const float*)d_in[2];   // (N, 3) f32
    const float* adj   = (const float*)d_in[3];   // (N, N) f32
    float* out = (float*)d_out;                   // scalar f32

    // Workspace: p[N] | partAcc[CHUNKS*N] | partAdj[CHUNKS*N]  (~1.6 MB)
    float* p       = (float*)d_ws;
    float* partAcc = p + N_VERT;
    float* partAdj = partAcc + (size_t)CHUNKS * N_VERT;

    logp_gather_kernel<<<N_VERT / WAVES_PER_BLOCK, 256, 0, stream>>>(pred, gt, p);

    dim3 grid(N_VERT / TILE, CHUNKS);             // 768 x 16 blocks
    pair_loss_kernel<<<grid, 256, 0, stream>>>(verts, adj, p, partAcc, partAdj);

    final_reduce_kernel<<<1, 256, 0, stream>>>(partAcc, partAdj, out);
}